// GNNModel_87703232184477
// MI455X (gfx1250) — compile-verified
//
#include <hip/hip_runtime.h>
#include <hip/hip_bf16.h>
#include <math.h>

typedef __attribute__((ext_vector_type(2))) float v2f;
typedef __attribute__((ext_vector_type(8))) float v8f;

#define IN_F   128
#define HID_F  128
#define OUT_F  16
#define N_GRAPHS 256

// ---------------------------------------------------------------------------
// 0) zero-init a float region (agg + deg are contiguous in ws)
// ---------------------------------------------------------------------------
__global__ void zero_kernel(float* __restrict__ p, size_t count) {
    size_t i = (size_t)blockIdx.x * blockDim.x + threadIdx.x;
    size_t stride = (size_t)gridDim.x * blockDim.x;
    for (; i < count; i += stride) p[i] = 0.0f;
}

// ---------------------------------------------------------------------------
// 1) degree count: deg[dst] += 1 per edge (self loop folded in later as +1)
// ---------------------------------------------------------------------------
__global__ void deg_kernel(const int* __restrict__ dst, float* __restrict__ deg, int E) {
    int e = blockIdx.x * blockDim.x + threadIdx.x;
    if (e < E) atomicAdd(&deg[dst[e]], 1.0f);
}

// 2) deg -> (deg+1)^{-1/2} in place (self loop guarantees deg+1 >= 1)
__global__ void dinv_kernel(float* __restrict__ deg, int N) {
    int i = blockIdx.x * blockDim.x + threadIdx.x;
    if (i < N) deg[i] = rsqrtf(deg[i] + 1.0f);
}

// ---------------------------------------------------------------------------
// 3) h = x @ W1   (bias added post-aggregation per GCNConv)
//    WMMA f32 16x16x4. 256 threads = 8 waves; block does 128 rows x 128 cols.
//    W1 (64 KB) staged in LDS once per workgroup (320 KB/WGP available).
// ---------------------------------------------------------------------------
__global__ void __launch_bounds__(256)
gemm_xw1_kernel(const float* __restrict__ x, const float* __restrict__ W1,
                float* __restrict__ h, int N) {
    __shared__ float sW[IN_F * HID_F];  // 64 KB

    // cooperative load of W1 into LDS (float4)
    {
        const float4* Wv = (const float4*)W1;
        float4* sWv = (float4*)sW;
        for (int i = threadIdx.x; i < IN_F * HID_F / 4; i += 256) sWv[i] = Wv[i];
    }
    __syncthreads();

    const int wave = threadIdx.x >> 5;
    const int lane = threadIdx.x & 31;
    const int row0 = blockIdx.x * 128 + wave * 16;

    // A-fragment addressing (32-bit A 16x4 layout):
    //   lanes 0-15: M=lane,    K pair = {k0, k0+1}
    //   lanes16-31: M=lane-16, K pair = {k0+2, k0+3}
    const int mrow  = row0 + (lane & 15);
    const int khalf = (lane >> 4) << 1;
    const int arow  = (mrow < N) ? mrow : (N - 1);       // clamp; masked at store
    const float* xr = x + (size_t)arow * IN_F;

    v8f zero = {};
    v8f acc[8];
#pragma unroll
    for (int t = 0; t < 8; ++t) acc[t] = zero;

    const int ncol = lane & 15;
    for (int k0 = 0; k0 < IN_F; k0 += 4) {
        v2f a;
        a.x = xr[k0 + khalf];
        a.y = xr[k0 + khalf + 1];
#pragma unroll
        for (int t = 0; t < 8; ++t) {
            const int col = t * 16 + ncol;
            v2f b;
            b.x = sW[(k0 + khalf)     * HID_F + col];
            b.y = sW[(k0 + khalf + 1) * HID_F + col];
            acc[t] = __builtin_amdgcn_wmma_f32_16x16x4_f32(
                false, a, false, b, (short)0, acc[t], false, false);
        }
    }

    // C/D layout: VGPR v holds (M = v + 8*(lane>=16), N = lane&15)
    const int mhi = (lane >> 4) << 3;
    if (row0 + 16 <= N) {
        // fast path: full tile, no per-element guard (all blocks but the last)
#pragma unroll
        for (int t = 0; t < 8; ++t) {
#pragma unroll
            for (int v = 0; v < 8; ++v) {
                h[(size_t)(row0 + v + mhi) * HID_F + t * 16 + ncol] = acc[t][v];
            }
        }
    } else {
#pragma unroll
        for (int t = 0; t < 8; ++t) {
#pragma unroll
            for (int v = 0; v < 8; ++v) {
                const int r = row0 + v + mhi;
                if (r < N) h[(size_t)r * HID_F + t * 16 + ncol] = acc[t][v];
            }
        }
    }
}

// ---------------------------------------------------------------------------
// 4) scatter: one wave per work item (E edges + N self loops)
//    out[dst] += dinv[src]*dinv[dst] * h[src];  lane covers 4 features
// ---------------------------------------------------------------------------
__global__ void scatter_kernel(const float* __restrict__ h, const float* __restrict__ dinv,
                               const int* __restrict__ src, const int* __restrict__ dst,
                               float* __restrict__ out, int E, int N) {
    const int item = (int)(((size_t)blockIdx.x * blockDim.x + threadIdx.x) >> 5);
    const int lane = threadIdx.x & 31;
    const int total = E + N;
    if (item >= total) return;

    int s, d;
    if (item < E) { s = src[item]; d = dst[item]; }
    else          { s = d = item - E; }

    const float norm = dinv[s] * dinv[d];
    const float4 v = ((const float4*)(h + (size_t)s * HID_F))[lane];
    float* o = out + (size_t)d * HID_F + lane * 4;
    atomicAdd(o + 0, v.x * norm);
    atomicAdd(o + 1, v.y * norm);
    atomicAdd(o + 2, v.z * norm);
    atomicAdd(o + 3, v.w * norm);
}

// ---------------------------------------------------------------------------
// 5) relu(agg + b1) + segment-max pool, atomic-free.
//    batch[n] = floor(n*256/N) is sorted, so graph g owns the contiguous
//    node range [ceil(g*N/G), ceil((g+1)*N/G)). One block per graph,
//    one thread per feature; column-streaming max reduction in registers.
//    max(relu(v)) == max(0, max(v)) since relu is monotone -> init m = 0.
// ---------------------------------------------------------------------------
__global__ void __launch_bounds__(HID_F)
pool_kernel(const float* __restrict__ agg, const float* __restrict__ b1,
            float* __restrict__ pooled, int N) {
    const int g = blockIdx.x;
    const int f = threadIdx.x;
    const int nstart = (int)(((long long)g       * N + N_GRAPHS - 1) / N_GRAPHS);
    const int nend   = (int)(((long long)(g + 1) * N + N_GRAPHS - 1) / N_GRAPHS);
    const float bias = b1[f];
    float m = 0.0f;
    for (int n = nstart; n < nend; ++n) {
        m = fmaxf(m, agg[(size_t)n * HID_F + f] + bias);
    }
    pooled[(size_t)g * HID_F + f] = m;
}

// ---------------------------------------------------------------------------
// 6) head: logits = pooled @ W2 + b2, then log_softmax. One thread per graph.
// ---------------------------------------------------------------------------
__global__ void head_kernel(const float* __restrict__ pooled, const float* __restrict__ W2,
                            const float* __restrict__ b2, float* __restrict__ out) {
    const int g = blockIdx.x * blockDim.x + threadIdx.x;
    if (g >= N_GRAPHS) return;

    float l[OUT_F];
#pragma unroll
    for (int j = 0; j < OUT_F; ++j) l[j] = b2[j];

    const float* p = pooled + (size_t)g * HID_F;
    for (int k = 0; k < HID_F; ++k) {
        const float pv = p[k];
        const float* wrow = W2 + (size_t)k * OUT_F;
#pragma unroll
        for (int j = 0; j < OUT_F; ++j) l[j] += pv * wrow[j];
    }

    float m = l[0];
#pragma unroll
    for (int j = 1; j < OUT_F; ++j) m = fmaxf(m, l[j]);
    float s = 0.0f;
#pragma unroll
    for (int j = 0; j < OUT_F; ++j) s += __expf(l[j] - m);
    const float lse = __logf(s);
#pragma unroll
    for (int j = 0; j < OUT_F; ++j) out[(size_t)g * OUT_F + j] = l[j] - m - lse;
}

// ---------------------------------------------------------------------------
extern "C" void kernel_launch(void* const* d_in, const int* in_sizes, int n_in,
                              void* d_out, int out_size, void* d_ws, size_t ws_size,
                              hipStream_t stream) {
    const float* x  = (const float*)d_in[0];
    const float* W1 = (const float*)d_in[1];
    const float* b1 = (const float*)d_in[2];
    const float* W2 = (const float*)d_in[3];
    const float* b2 = (const float*)d_in[4];
    const int* edge_index = (const int*)d_in[5];

    const int N = in_sizes[0] / IN_F;
    const int E = in_sizes[5] / 2;
    const int* esrc = edge_index;       // edge_index[0]
    const int* edst = edge_index + E;   // edge_index[1]

    // workspace layout (contiguous): h | agg | deg(->dinv) | pooled
    float* ws     = (float*)d_ws;
    float* h      = ws;
    float* agg    = h   + (size_t)N * HID_F;
    float* deg    = agg + (size_t)N * HID_F;
    float* pooled = deg + (size_t)N;

    // 0) zero agg + deg (contiguous region starting at agg)
    const size_t zcount = (size_t)N * HID_F + (size_t)N;
    zero_kernel<<<2048, 256, 0, stream>>>(agg, zcount);

    // 1) degree
    deg_kernel<<<(E + 255) / 256, 256, 0, stream>>>(edst, deg, E);

    // 2) dinv
    dinv_kernel<<<(N + 255) / 256, 256, 0, stream>>>(deg, N);

    // 3) h = x @ W1 (WMMA)
    gemm_xw1_kernel<<<(N + 127) / 128, 256, 0, stream>>>(x, W1, h, N);

    // 4) edge + self-loop scatter (one wave32 per item)
    {
        const long long items = (long long)E + N;
        const long long blocks = (items + 7) / 8;  // 8 waves / 256-thread block
        scatter_kernel<<<(int)blocks, 256, 0, stream>>>(h, deg, esrc, edst, agg, E, N);
    }

    // 5) relu + bias + segment-max pool (atomic-free, contiguous segments)
    pool_kernel<<<N_GRAPHS, HID_F, 0, stream>>>(agg, b1, pooled, N);

    // 6) head GEMM + log_softmax
    head_kernel<<<1, N_GRAPHS, 0, stream>>>(pooled, W2, b2, (float*)d_out);
}